// adder2d_24472723653056
// MI455X (gfx1250) — compile-verified
//
#include <hip/hip_runtime.h>

// ---------------- problem constants (adder2d) ----------------
#define CN   16      // batch
#define CC   64      // in channels
#define CH   32      // height
#define CW   32      // width
#define CF   64      // out channels
#define CK   576     // C*3*3
#define TROWS 2      // output rows per block tile
#define SROWS 4      // staged input rows (TROWS + 2 halo), power of 2
#define RSTRIDE 40   // padded LDS row stride (floats); data at cc=4..35, pads at 3 and 36
#define NTHREADS 64  // 2 waves: one 2x32 pixel tile
#define TNTH 128     // threads for the transpose prep kernel
#define CCHUNK 8                     // channels per W chunk
#define WCHUNK (CCHUNK * 9 * CF)     // 4608 floats = 18432 B
#define WSTG   (WCHUNK / NTHREADS)   // 72 async b32 loads per thread per chunk

// ---------------- CDNA5 async global->LDS helpers ----------------
#if defined(__has_builtin)
#if __has_builtin(__builtin_amdgcn_global_load_async_to_lds_b32)
#define HAVE_ASYNC_LDS 1
#endif
#endif

__device__ __forceinline__ void async_copy_f32(const float* g, float* l) {
#ifdef HAVE_ASYNC_LDS
    typedef __attribute__((address_space(1))) int gint_t;   // global int*
    typedef __attribute__((address_space(3))) int lint_t;   // LDS int*
    __builtin_amdgcn_global_load_async_to_lds_b32(
        (gint_t*)(g), (lint_t*)(l), /*imm offset*/0, /*cpol*/0);
#else
    *l = *g;   // synchronous fallback; barriers below keep it correct
#endif
}

__device__ __forceinline__ void wait_async_all() {
#ifdef HAVE_ASYNC_LDS
#if __has_builtin(__builtin_amdgcn_s_wait_asynccnt)
    __builtin_amdgcn_s_wait_asynccnt(0);
#else
    asm volatile("s_wait_asynccnt 0" ::: "memory");
#endif
#endif
}

// ---------------- main adder2d kernel (first in file -> visible in disasm) ----
__global__ __launch_bounds__(NTHREADS)
void adder2d_main(const float* __restrict__ x,
                  const float* __restrict__ wt,
                  float* __restrict__ out) {
    // padded x tile: [64 ch][4 rows][40 cols] floats = 40960 B
    __shared__ float4 xtile4[(CC * SROWS * RSTRIDE) / 4];
    // double-buffered W chunk: 2 x 8ch x 9 x 64f = 2 x 18432 B
    __shared__ float4 wbuf4[2][WCHUNK / 4];
    float* xtile = (float*)xtile4;

    const int tid  = threadIdx.x;
    const int n    = blockIdx.x >> 4;       // image
    const int tile = blockIdx.x & 15;       // row-tile
    const int h0   = tile * TROWS;

    // ---- zero only the pad columns cc=3 and cc=36 (512 cells) ----
    #pragma unroll
    for (int s = 0; s < (CC * SROWS * 2) / NTHREADS; ++s) {   // 8 iters
        int p    = tid + NTHREADS * s;
        int side = p & 1;
        int rr   = (p >> 1) & (SROWS - 1);
        int c    = p >> 3;
        xtile[(c * SROWS + rr) * RSTRIDE + 3 + side * 33] = 0.f;
    }

    // ---- async-stage x interior; zero-store out-of-image halo rows ----
    const float* xn = x + (size_t)n * CC * CH * CW;
    for (int s = 0; s < (CC * SROWS * CW) / NTHREADS; ++s) {  // 128 iters
        int e   = tid + NTHREADS * s;
        int col = e & 31;
        int rr  = (e >> 5) & (SROWS - 1);
        int c   = e >> 7;
        int gh  = h0 - 1 + rr;                 // wave-uniform branch
        float* lp = &xtile[(c * SROWS + rr) * RSTRIDE + 4 + col];
        if (gh >= 0 && gh < CH) {
            const float* gp = xn + ((size_t)c * CH + gh) * CW + col;
            async_copy_f32(gp, lp);            // disjoint from zero-stores
        } else {
            *lp = 0.f;                         // halo row outside image
        }
    }
    // ---- async-stage W chunk 0 (coalesced from transposed WT[k][f]) ----
    {
        float* wdst = (float*)wbuf4[0];
        for (int s = 0; s < WSTG; ++s) {
            int e = tid + NTHREADS * s;
            async_copy_f32(wt + e, wdst + e);
        }
    }
    wait_async_all();   // async writes done (zero-stores fenced by barrier's dscnt)
    __syncthreads();

    // ---- compute: each thread owns one pixel, 64 filter accumulators ----
    const int r    = tid >> 5;   // row in tile  0..1
    const int wcol = tid & 31;   // col          0..31

    float acc[CF];
    #pragma unroll
    for (int f = 0; f < CF; ++f) acc[f] = 0.f;

    #pragma unroll 1
    for (int chk = 0; chk < CC / CCHUNK; ++chk) {
        // prefetch next W chunk while computing this one
        if (chk < CC / CCHUNK - 1) {
            const float* wsrc = wt + (size_t)(chk + 1) * WCHUNK;
            float*       wdst = (float*)wbuf4[(chk + 1) & 1];
            for (int s = 0; s < WSTG; ++s) {
                int e = tid + NTHREADS * s;
                async_copy_f32(wsrc + e, wdst + e);
            }
        }

        const float* wb = (const float*)wbuf4[chk & 1];
        #pragma unroll 1
        for (int cl = 0; cl < CCHUNK; ++cl) {
            const int c = chk * CCHUNK + cl;
            // 3x3 input patch for this pixel (LDS, consecutive banks per lane)
            float xv[9];
            #pragma unroll
            for (int i = 0; i < 3; ++i)
                #pragma unroll
                for (int j = 0; j < 3; ++j)
                    xv[i * 3 + j] =
                        xtile[(c * SROWS + (r + i)) * RSTRIDE + 3 + wcol + j];

            #pragma unroll
            for (int kk = 0; kk < 9; ++kk) {
                // ds_load_b128 broadcast, in-order & pipelined
                const float4* wp = (const float4*)(wb + (cl * 9 + kk) * CF);
                const float xk = xv[kk];
                #pragma unroll
                for (int f4 = 0; f4 < CF / 4; ++f4) {
                    float4 wv = wp[f4];
                    acc[f4 * 4 + 0] += __builtin_fabsf(wv.x - xk);
                    acc[f4 * 4 + 1] += __builtin_fabsf(wv.y - xk);
                    acc[f4 * 4 + 2] += __builtin_fabsf(wv.z - xk);
                    acc[f4 * 4 + 3] += __builtin_fabsf(wv.w - xk);
                }
            }
        }

        wait_async_all();   // next chunk fully in LDS (covered by ~9200 VALU ops)
        __syncthreads();    // all waves done reading old buffer / writing new one
    }

    // ---- write out[n][f][h0+r][wcol] = -acc[f] (coalesced over wcol) ----
    float* op = out + (((size_t)n * CF) * CH + (h0 + r)) * CW + wcol;
    #pragma unroll
    for (int f = 0; f < CF; ++f)
        op[(size_t)f * CH * CW] = -acc[f];
}

// ---------------- prep: W[f][k] -> WT[k][f] ----------------
__global__ void adder2d_transpose_w(const float* __restrict__ Wfk,
                                    float* __restrict__ wt) {
    int idx = blockIdx.x * TNTH + threadIdx.x;  // 36864 total
    int f = idx & (CF - 1);
    int k = idx >> 6;
    if (k < CK) wt[idx] = Wfk[f * CK + k];
}

// ---------------- launch ----------------
extern "C" void kernel_launch(void* const* d_in, const int* in_sizes, int n_in,
                              void* d_out, int out_size, void* d_ws, size_t ws_size,
                              hipStream_t stream) {
    const float* x = (const float*)d_in[0];   // [16,64,32,32] f32
    const float* W = (const float*)d_in[1];   // [64,64,3,3]   f32
    float* out = (float*)d_out;               // [16,64,32,32] f32
    float* wt  = (float*)d_ws;                // WT[k][f], 576*64*4 = 147456 B

    (void)in_sizes; (void)n_in; (void)out_size; (void)ws_size;

    // 1) transpose W -> WT[k][f]
    adder2d_transpose_w<<<(CK * CF) / TNTH, TNTH, 0, stream>>>(W, wt);

    // 2) main adder conv: 16 images x 16 two-row tiles = 256 blocks
    adder2d_main<<<CN * (CH / TROWS), NTHREADS, 0, stream>>>(x, wt, out);
}